// MBSRHGCN_19610820674331
// MI455X (gfx1250) — compile-verified
//
#include <hip/hip_runtime.h>
#include <hip/hip_bf16.h>

typedef __attribute__((ext_vector_type(2))) float v2f;
typedef __attribute__((ext_vector_type(8))) float v8f;
typedef __attribute__((ext_vector_type(4))) unsigned int u32x4;
typedef __attribute__((ext_vector_type(8))) int i32x8;
typedef __attribute__((ext_vector_type(4))) int i32x4;

#define NU_   80000
#define NS_   20000
#define NM_   30000
#define D_    128
#define B_    2048
#define L_    32

#if __has_builtin(__builtin_amdgcn_tensor_load_to_lds) && \
    __has_builtin(__builtin_amdgcn_s_wait_tensorcnt)
#define USE_TDM 1
#else
#define USE_TDM 0
#endif

// ---------------------------------------------------------------------------
// Elementwise helpers
// ---------------------------------------------------------------------------
__global__ void zero_kernel(float* __restrict__ p, long n) {
    long i = (long)blockIdx.x * blockDim.x + threadIdx.x;
    if (i < n) p[i] = 0.f;
}

// p[i] *= d[i / 128]   (row-wise d_inv scaling)
__global__ void scale_rows_kernel(float* __restrict__ p, const float* __restrict__ d, long n) {
    long i = (long)blockIdx.x * blockDim.x + threadIdx.x;
    if (i < n) p[i] *= d[i >> 7];
}

// P[i] = (h0[i] + P[i] + Q[i]) / 3, h0 virtually concatenated from a0 (i<split) and a1.
__global__ void avg3_kernel(const float* __restrict__ a0, const float* __restrict__ a1,
                            long split, float* __restrict__ P, const float* __restrict__ Q,
                            long n) {
    long i = (long)blockIdx.x * blockDim.x + threadIdx.x;
    if (i < n) {
        float h0 = (i < split) ? a0[i] : a1[i - split];
        P[i] = (h0 + P[i] + Q[i]) * (1.f / 3.f);
    }
}

// ---------------------------------------------------------------------------
// Scatter-add spMM: out[rows[e],:] += vals[e] * x[cols[e],:]
// one wave per edge, float4 per lane (D = 128 = 32 lanes * 4).
// Edge lists are streamed once -> non-temporal loads so the embedding tables
// (the reused gather/scatter working set) stay resident in the 192MB L2.
// x is virtual concat: col < split -> x0[col], else x1[col-split].
// ---------------------------------------------------------------------------
__global__ void spmm_kernel(const int* __restrict__ rows, const int* __restrict__ cols,
                            const float* __restrict__ vals,
                            const float* __restrict__ x0, const float* __restrict__ x1,
                            int split, float* __restrict__ out, int nE) {
    long t = (long)blockIdx.x * blockDim.x + threadIdx.x;
    int e = (int)(t >> 5);
    int lane = (int)(t & 31);
    if (e >= nE) return;
    int r = __builtin_nontemporal_load(rows + e);
    int c = __builtin_nontemporal_load(cols + e);
    float v = __builtin_nontemporal_load(vals + e);
    const float* src = (c < split) ? (x0 + (long)c * D_) : (x1 + (long)(c - split) * D_);
    const float4 x = *(const float4*)(src + lane * 4);
    float* dst = out + (long)r * D_ + lane * 4;
    atomicAdd(dst + 0, v * x.x);
    atomicAdd(dst + 1, v * x.y);
    atomicAdd(dst + 2, v * x.z);
    atomicAdd(dst + 3, v * x.w);
}

// ---------------------------------------------------------------------------
// Attention score MLP via V_WMMA_F32_16X16X4_F32.
// A tile = 16 rows of at_emb (B*L, 256), B = att_w1 (256x16) staged in LDS
// via the Tensor Data Mover (one contiguous 16KB D# tile, wave 0 issues,
// s_wait_tensorcnt 0, workgroup barrier releases the other 7 waves).
// Epilogue: relu(+b1), dot with att_w2 (16) via half-wave shfl reduction, +b2.
// One wave per 16-row tile; 8 waves/block; grid = 4096/8 = 512 full blocks.
// ---------------------------------------------------------------------------
__global__ __launch_bounds__(256)
void attn_score_kernel(const int* __restrict__ member_masked,
                       const int* __restrict__ service_inputs,
                       const float* __restrict__ uiE,      // final (NU+NS, 128)
                       const float* __restrict__ w1,       // (256,16)
                       const float* __restrict__ b1,       // (16)
                       const float* __restrict__ w2,       // (16,1)
                       const float* __restrict__ b2,       // (1)
                       float* __restrict__ scores)         // (B*L)
{
    __shared__ float lw1[256 * 16];
    int tid = threadIdx.x;
#if USE_TDM
    if (tid < 32) {
        // D# group0: count=1, lds_addr, 57-bit global addr, type=2 ("image")
        unsigned long long ga = (unsigned long long)(const void*)w1;
        unsigned lds = (unsigned)(unsigned long long)(void*)lw1;
        u32x4 g0;
        g0.x = 1u;                                            // count=1, user desc
        g0.y = lds;                                           // lds_addr (bytes)
        g0.z = (unsigned)(ga & 0xFFFFFFFFu);                  // global_addr[31:0]
        g0.w = (unsigned)((ga >> 32) & 0x01FFFFFFu) | (2u << 30); // [56:32] | type=2
        // D# group1: data_size=4B; tensor_dim0=4096, tensor_dim1=1;
        //            tile_dim0=4096, tile_dim1=1; tensor_dim0_stride=4096
        i32x8 g1;
        g1[0] = (int)(2u << 16);          // workgroup_mask=0, data_size=2 (4B)
        g1[1] = (int)(4096u << 16);       // tensor_dim0[15:0] in [31:16]
        g1[2] = (int)(1u << 16);          // tensor_dim0[31:16]=0 | tensor_dim1[15:0]=1
        g1[3] = (int)(4096u << 16);       // tensor_dim1[31:16]=0 | tile_dim0=4096
        g1[4] = 1;                        // tile_dim1=1, tile_dim2=0
        g1[5] = 4096;                     // tensor_dim0_stride[31:0]
        g1[6] = 0;
        g1[7] = 0;
        i32x4 gz = {0, 0, 0, 0};
#if defined(__clang_major__) && (__clang_major__ >= 23)
        i32x8 gz8 = {0, 0, 0, 0, 0, 0, 0, 0};
        __builtin_amdgcn_tensor_load_to_lds(g0, g1, gz, gz, gz8, 0);
#else
        __builtin_amdgcn_tensor_load_to_lds(g0, g1, gz, gz, 0);
#endif
        __builtin_amdgcn_s_wait_tensorcnt(0);
    }
#else
    for (int i = tid; i < 256 * 16; i += 256) lw1[i] = w1[i];
#endif
    __syncthreads();

    int wave = tid >> 5;
    int lane = tid & 31;
    int tile = blockIdx.x * 8 + wave;          // [0, 4096)
    int r = tile * 16 + (lane & 15);           // flat row in [0, B*L)
    int b = r >> 5;                            // / L
    int l = r & 31;

    int uid = member_masked[b * L_ + l];
    int sid = service_inputs[b];
    const float* mrow = uiE + (long)uid * D_;
    const float* srow = uiE + (long)(NU_ + sid) * D_;

    int koff = (lane >> 4) * 2;                // lanes 0-15: K=0,1 ; lanes 16-31: K=2,3
    int n = lane & 15;

    v8f c = {};
    #pragma unroll 8
    for (int kb = 0; kb < 256; kb += 4) {
        int k0 = kb + koff;                    // k0 even -> pair stays in one half
        const float* src = (k0 < 128) ? (mrow + k0) : (srow + (k0 - 128));
        v2f a, bb;
        a.x = src[0];
        a.y = src[1];
        bb.x = lw1[k0 * 16 + n];
        bb.y = lw1[(k0 + 1) * 16 + n];
        c = __builtin_amdgcn_wmma_f32_16x16x4_f32(false, a, false, bb, (short)0, c,
                                                  false, false);
    }

    float bias = b1[n];
    float wn = w2[n];
    float b2v = b2[0];
    #pragma unroll
    for (int rr = 0; rr < 8; ++rr) {
        float v = c[rr] + bias;
        v = v > 0.f ? v : 0.f;
        float t = v * wn;
        #pragma unroll
        for (int off = 1; off < 16; off <<= 1) t += __shfl_xor(t, off, 16);
        int row = (lane < 16) ? rr : (rr + 8);
        if (n == 0) scores[tile * 16 + row] = t + b2v;
    }
}

// ---------------------------------------------------------------------------
// Masked softmax over L=32 + attention-weighted member sum + mashup gather.
// One wave per batch element.
// ---------------------------------------------------------------------------
__global__ __launch_bounds__(32)
void attn_apply_kernel(const float* __restrict__ scores, const float* __restrict__ mask,
                       const int* __restrict__ member_masked,
                       const int* __restrict__ mashup_inputs,
                       const float* __restrict__ uiE, const float* __restrict__ mE,
                       float* __restrict__ mashup_emb)    // (B,128)
{
    int b = blockIdx.x;
    int lane = threadIdx.x;
    float s = scores[b * L_ + lane];
    if (mask[b * L_ + lane] > 0.f) s = -1.0e9f;
    float mx = s;
    #pragma unroll
    for (int off = 1; off < 32; off <<= 1) mx = fmaxf(mx, __shfl_xor(mx, off, 32));
    float e = __expf(s - mx);
    float sum = e;
    #pragma unroll
    for (int off = 1; off < 32; off <<= 1) sum += __shfl_xor(sum, off, 32);
    float w = e / sum;

    int uid = member_masked[b * L_ + lane];
    int mid = mashup_inputs[b];
    #pragma unroll
    for (int chunk = 0; chunk < 4; ++chunk) {
        int cidx = chunk * 32 + lane;
        float acc = 0.f;
        for (int l = 0; l < 32; ++l) {
            float wl = __shfl(w, l, 32);
            int ul = __shfl(uid, l, 32);
            acc += wl * uiE[(long)ul * D_ + cidx];
        }
        mashup_emb[(long)b * D_ + cidx] = acc + mE[(long)mid * D_ + cidx];
    }
}

// ---------------------------------------------------------------------------
// Prediction MLP via WMMA. new_emb = [me*se, me, se] (2048, 384), W1 (384,8)
// zero-padded to 16 cols in LDS (cooperative load: TDM cannot zero-fill),
// then relu / dot with pred_w2 / sigmoid.
// 128 tiles -> 16 blocks x 8 waves, all waves full.
// ---------------------------------------------------------------------------
__global__ __launch_bounds__(256)
void pred_kernel(const int* __restrict__ service_inputs,
                 const float* __restrict__ uiE,
                 const float* __restrict__ mashup_emb,    // (B,128)
                 const float* __restrict__ pw1,           // (384,8)
                 const float* __restrict__ pb1,           // (8)
                 const float* __restrict__ pw2,           // (8,1)
                 const float* __restrict__ pb2,           // (1)
                 float* __restrict__ y)                   // (B)
{
    __shared__ float lw1[384 * 16];
    int tid = threadIdx.x;
    for (int i = tid; i < 384 * 16; i += 256) {
        int k = i >> 4, n = i & 15;
        lw1[i] = (n < 8) ? pw1[k * 8 + n] : 0.f;
    }
    __syncthreads();

    int wave = tid >> 5;
    int lane = tid & 31;
    int tile = blockIdx.x * 8 + wave;          // [0, 128)
    int r = tile * 16 + (lane & 15);           // batch row
    int sid = service_inputs[r];
    const float* me = mashup_emb + (long)r * D_;
    const float* se = uiE + (long)(NU_ + sid) * D_;

    int koff = (lane >> 4) * 2;
    int n = lane & 15;

    v8f c = {};
    #pragma unroll 8
    for (int kb = 0; kb < 384; kb += 4) {
        int k0 = kb + koff;
        float a0, a1;
        if (k0 < 128)      { a0 = me[k0] * se[k0];     a1 = me[k0 + 1] * se[k0 + 1]; }
        else if (k0 < 256) { a0 = me[k0 - 128];        a1 = me[k0 - 127]; }
        else               { a0 = se[k0 - 256];        a1 = se[k0 - 255]; }
        v2f a, bb;
        a.x = a0; a.y = a1;
        bb.x = lw1[k0 * 16 + n];
        bb.y = lw1[(k0 + 1) * 16 + n];
        c = __builtin_amdgcn_wmma_f32_16x16x4_f32(false, a, false, bb, (short)0, c,
                                                  false, false);
    }

    float bias = (n < 8) ? pb1[n] : 0.f;
    float wn   = (n < 8) ? pw2[n] : 0.f;
    float b2v = pb2[0];
    #pragma unroll
    for (int rr = 0; rr < 8; ++rr) {
        float v = c[rr] + bias;
        v = v > 0.f ? v : 0.f;
        float t = v * wn;
        #pragma unroll
        for (int off = 1; off < 16; off <<= 1) t += __shfl_xor(t, off, 16);
        int row = (lane < 16) ? rr : (rr + 8);
        if (n == 0) y[tile * 16 + row] = 1.f / (1.f + __expf(-(t + b2v)));
    }
}

// ---------------------------------------------------------------------------
extern "C" void kernel_launch(void* const* d_in, const int* in_sizes, int n_in,
                              void* d_out, int out_size, void* d_ws, size_t ws_size,
                              hipStream_t stream) {
    const int*   mashup_inputs  = (const int*)d_in[0];
    const int*   service_inputs = (const int*)d_in[1];
    const int*   member_masked  = (const int*)d_in[2];
    const float* mask           = (const float*)d_in[3];
    const int*   adj_rows       = (const int*)d_in[4];
    const int*   adj_cols       = (const int*)d_in[5];
    const float* adj_vals       = (const float*)d_in[6];
    const int*   A_rows         = (const int*)d_in[7];
    const int*   A_cols         = (const int*)d_in[8];
    const float* A_vals         = (const float*)d_in[9];
    const float* d_inv          = (const float*)d_in[10];
    const float* user_tbl       = (const float*)d_in[11];
    const float* service_tbl    = (const float*)d_in[12];
    const float* mashup_tbl     = (const float*)d_in[13];
    const float* att_w1         = (const float*)d_in[14];
    const float* att_b1         = (const float*)d_in[15];
    const float* att_w2         = (const float*)d_in[16];
    const float* att_b2         = (const float*)d_in[17];
    const float* pred_w1        = (const float*)d_in[18];
    const float* pred_b1        = (const float*)d_in[19];
    const float* pred_w2        = (const float*)d_in[20];
    const float* pred_b2        = (const float*)d_in[21];
    float* y = (float*)d_out;

    const int nE_ui = in_sizes[4];
    const int nE_mm = in_sizes[7];

    const long N_UI = (long)(NU_ + NS_) * D_;  // 12.8M floats
    const long N_M  = (long)NM_ * D_;          // 3.84M floats

    float* P   = (float*)d_ws;                 // final ui embeddings live here
    float* Q   = P + N_UI;
    float* Pm  = Q;                            // mashup buffers overlay Q (used after ui done)
    float* Qm  = Q + N_M;
    float* scores     = Q + N_UI;              // B*L
    float* mashup_emb = scores + (long)B_ * L_;// B*128

    const int T = 256;
    auto blocks = [](long n, int t) { return (unsigned)((n + t - 1) / t); };

    // ---- user/service graph: 2 LightGCN layers + average ----
    zero_kernel<<<blocks(N_UI, T), T, 0, stream>>>(P, N_UI);
    spmm_kernel<<<blocks((long)nE_ui * 32, T), T, 0, stream>>>(
        adj_rows, adj_cols, adj_vals, user_tbl, service_tbl, NU_, P, nE_ui);
    zero_kernel<<<blocks(N_UI, T), T, 0, stream>>>(Q, N_UI);
    spmm_kernel<<<blocks((long)nE_ui * 32, T), T, 0, stream>>>(
        adj_rows, adj_cols, adj_vals, P, P, NU_ + NS_ + 1, Q, nE_ui);
    avg3_kernel<<<blocks(N_UI, T), T, 0, stream>>>(
        user_tbl, service_tbl, (long)NU_ * D_, P, Q, N_UI);

    // ---- mashup graph: 2 layers with d_inv scaling + average ----
    zero_kernel<<<blocks(N_M, T), T, 0, stream>>>(Pm, N_M);
    spmm_kernel<<<blocks((long)nE_mm * 32, T), T, 0, stream>>>(
        A_rows, A_cols, A_vals, mashup_tbl, mashup_tbl, NM_ + 1, Pm, nE_mm);
    scale_rows_kernel<<<blocks(N_M, T), T, 0, stream>>>(Pm, d_inv, N_M);
    zero_kernel<<<blocks(N_M, T), T, 0, stream>>>(Qm, N_M);
    spmm_kernel<<<blocks((long)nE_mm * 32, T), T, 0, stream>>>(
        A_rows, A_cols, A_vals, Pm, Pm, NM_ + 1, Qm, nE_mm);
    scale_rows_kernel<<<blocks(N_M, T), T, 0, stream>>>(Qm, d_inv, N_M);
    avg3_kernel<<<blocks(N_M, T), T, 0, stream>>>(
        mashup_tbl, mashup_tbl, N_M, Pm, Qm, N_M);

    // ---- attention MLP (WMMA, TDM-staged W1) -> scores ----
    attn_score_kernel<<<(B_ * L_ / 16) / 8, 256, 0, stream>>>(
        member_masked, service_inputs, P, att_w1, att_b1, att_w2, att_b2, scores);

    // ---- masked softmax + weighted member sum + mashup gather ----
    attn_apply_kernel<<<B_, 32, 0, stream>>>(
        scores, mask, member_masked, mashup_inputs, P, Pm, mashup_emb);

    // ---- prediction MLP (WMMA) -> sigmoid -> y ----
    pred_kernel<<<(B_ / 16) / 8, 256, 0, stream>>>(
        service_inputs, P, mashup_emb, pred_w1, pred_b1, pred_w2, pred_b2, y);
}